// PriorNetwork_9818295239116
// MI455X (gfx1250) — compile-verified
//
#include <hip/hip_runtime.h>

typedef __attribute__((ext_vector_type(2))) float v2f;
typedef __attribute__((ext_vector_type(8))) float v8f;

#define WMMA_F32(a, b, c) \
  __builtin_amdgcn_wmma_f32_16x16x4_f32(false, (a), false, (b), (short)0, (c), false, false)

// Problem constants (match reference)
#define BB      1024
#define NN      100000
#define DD      64
#define HH      512
#define RTILES  4                          // M-tiles per wave in nn_kernel
#define NSPLIT  64
#define CHUNKS  (NN / 16)                  // 6250
#define CPS     ((CHUNKS + NSPLIT - 1) / NSPLIT)  // 98

// ---------------------------------------------------------------------------
// Kernel 1: per-row squared norms of the table (N floats)
// ---------------------------------------------------------------------------
__global__ void tnorm_kernel(const float* __restrict__ table,
                             float* __restrict__ tnorm) {
  int j = blockIdx.x * blockDim.x + threadIdx.x;
  if (j >= NN) return;
  const float4* row = (const float4*)(table + (size_t)j * DD);
  float s = 0.f;
#pragma unroll
  for (int i = 0; i < DD / 4; ++i) {
    float4 v = row[i];
    s += v.x * v.x + v.y * v.y + v.z * v.z + v.w * v.w;
  }
  tnorm[j] = s;
}

// ---------------------------------------------------------------------------
// Kernel 2: argmin of (||t||^2 - 2 c.t) via V_WMMA_F32_16X16X4_F32.
// grid = (16 M-groups of 4 tiles, NSPLIT N-splits), block = 1 wave.
// Each wave keeps 4 M-tiles of A (scaled by -2) in registers, so every B
// fragment loaded from L2 feeds 4 WMMA chains: 32 flop/byte from L2.
// A fragment: lane holds A[lane&15, kb + (lane>>4)*2 + {0,1}]   (16x4 f32)
// B fragment: lane holds B[kb + (lane>>4)*2 + {0,1}, lane&15] = T[lane&15, k]
// C/D: lane column n = lane&15, rows m = v + 8*(lane>>4), v = 0..7
// ---------------------------------------------------------------------------
__global__ __launch_bounds__(32) void nn_kernel(
    const float* __restrict__ codes, const float* __restrict__ table,
    const float* __restrict__ tnorm, float* __restrict__ pval,
    int* __restrict__ pidx) {
  const int lane = threadIdx.x;
  const int mgroup = blockIdx.x;  // group of RTILES M-tiles (4*16 batch rows)
  const int split = blockIdx.y;   // N split
  const int m = lane & 15;
  const int koff = (lane >> 4) * 2;

  // A = -2 * codes tiles, resident in registers for the whole N sweep.
  v2f a[RTILES][16];
#pragma unroll
  for (int t = 0; t < RTILES; ++t) {
    const float* crow =
        codes + (size_t)((mgroup * RTILES + t) * 16 + m) * DD + koff;
#pragma unroll
    for (int kb = 0; kb < 16; ++kb) {
      v2f v = *(const v2f*)(crow + kb * 4);
      a[t][kb].x = -2.0f * v.x;
      a[t][kb].y = -2.0f * v.y;
    }
  }

  float bestv[RTILES][8];
  int besti[RTILES][8];
#pragma unroll
  for (int t = 0; t < RTILES; ++t)
#pragma unroll
    for (int v = 0; v < 8; ++v) { bestv[t][v] = 3.4e38f; besti[t][v] = 0; }

  const int c0 = split * CPS;
  const int c1 = (c0 + CPS < CHUNKS) ? (c0 + CPS) : CHUNKS;

  for (int c = c0; c < c1; ++c) {
    const int col = c * 16 + m;  // this lane's global column (table row index)
    const float* trow = table + (size_t)col * DD + koff;

    // Prefetch next chunk's row for this lane (global_prefetch_b8).
    if (c + 1 < c1)
      __builtin_prefetch(table + (size_t)(col + 16) * DD + koff, 0, 1);

    v2f b[16];
#pragma unroll
    for (int kb = 0; kb < 16; ++kb) b[kb] = *(const v2f*)(trow + kb * 4);

    v8f acc[RTILES];
#pragma unroll
    for (int t = 0; t < RTILES; ++t) acc[t] = (v8f){};

#pragma unroll
    for (int kb = 0; kb < 16; ++kb)
#pragma unroll
      for (int t = 0; t < RTILES; ++t)  // 4 independent chains per B fragment
        acc[t] = WMMA_F32(a[t][kb], b[kb], acc[t]);

    const float tn = tnorm[col];
#pragma unroll
    for (int t = 0; t < RTILES; ++t)
#pragma unroll
      for (int v = 0; v < 8; ++v) {
        float s = tn + acc[t][v];  // ||t||^2 - 2 c.t (argmin-equivalent)
        if (s < bestv[t][v]) { bestv[t][v] = s; besti[t][v] = col; }
      }
  }

  // Min-reduce across the 16 lanes of each half (columns), keep indices.
#pragma unroll
  for (int off = 1; off < 16; off <<= 1) {
#pragma unroll
    for (int t = 0; t < RTILES; ++t)
#pragma unroll
      for (int v = 0; v < 8; ++v) {
        float ov = __shfl_xor(bestv[t][v], off, 32);
        int oi = __shfl_xor(besti[t][v], off, 32);
        if (ov < bestv[t][v]) { bestv[t][v] = ov; besti[t][v] = oi; }
      }
  }

  if ((lane & 15) == 0) {
    const int rbase = (lane >> 4) * 8;
#pragma unroll
    for (int t = 0; t < RTILES; ++t) {
      const int tile = mgroup * RTILES + t;
      const int out = (tile * NSPLIT + split) * 16 + rbase;
#pragma unroll
      for (int v = 0; v < 8; ++v) {
        pval[out + v] = bestv[t][v];
        pidx[out + v] = besti[t][v];
      }
    }
  }
}

// ---------------------------------------------------------------------------
// Kernel 3: reduce per-split partial argmins -> nn_idx[b]
// ---------------------------------------------------------------------------
__global__ void reduce_kernel(const float* __restrict__ pval,
                              const int* __restrict__ pidx,
                              int* __restrict__ nn_idx) {
  int b = blockIdx.x * blockDim.x + threadIdx.x;
  if (b >= BB) return;
  const int tile = b >> 4, r = b & 15;
  float bv = 3.4e38f;
  int bi = 0;
  for (int s = 0; s < NSPLIT; ++s) {
    float v = pval[(tile * NSPLIT + s) * 16 + r];
    if (v < bv) { bv = v; bi = pidx[(tile * NSPLIT + s) * 16 + r]; }
  }
  nn_idx[b] = bi;
}

// ---------------------------------------------------------------------------
// Kernel 4: h1 = relu(gather(table, nn_idx) @ W1 + b1)  via WMMA f32 16x16x4
// grid = (32 N-tiles of H, 64 M-tiles), block = 1 wave
// ---------------------------------------------------------------------------
__global__ __launch_bounds__(32) void h1_kernel(
    const float* __restrict__ table, const int* __restrict__ nn_idx,
    const float* __restrict__ W1, const float* __restrict__ b1,
    float* __restrict__ h1) {
  const int lane = threadIdx.x;
  const int ntile = blockIdx.x;  // 0..31
  const int mtile = blockIdx.y;  // 0..63
  const int n = lane & 15;       // A row == C column for this lane
  const int koff = (lane >> 4) * 2;
  const int nc = ntile * 16 + n;

  const int row = nn_idx[mtile * 16 + n];
  const float* prow = table + (size_t)row * DD + koff;

  const float bias = b1[nc];
  v8f acc;
#pragma unroll
  for (int v = 0; v < 8; ++v) acc[v] = bias;

#pragma unroll
  for (int kb = 0; kb < 16; ++kb) {
    v2f a = *(const v2f*)(prow + kb * 4);
    v2f b;
    b.x = W1[(size_t)(kb * 4 + koff + 0) * HH + nc];
    b.y = W1[(size_t)(kb * 4 + koff + 1) * HH + nc];
    acc = WMMA_F32(a, b, acc);
  }

#pragma unroll
  for (int v = 0; v < 8; ++v) {
    const int m = v + (lane >> 4) * 8;
    float h = acc[v];
    h1[(size_t)(mtile * 16 + m) * HH + nc] = h > 0.f ? h : 0.f;
  }
}

// ---------------------------------------------------------------------------
// Kernel 5: mu / logstd = h1 @ {Wu,Ws} + {bu,bs}  via WMMA f32 16x16x4
// grid = (4 N-tiles of D, 64 M-tiles, 2 heads), block = 1 wave
// ---------------------------------------------------------------------------
__global__ __launch_bounds__(32) void out_kernel(
    const float* __restrict__ h1, const float* __restrict__ Wu,
    const float* __restrict__ bu, const float* __restrict__ Ws,
    const float* __restrict__ bs, float* __restrict__ out) {
  const int lane = threadIdx.x;
  const int ntile = blockIdx.x;  // 0..3
  const int mtile = blockIdx.y;  // 0..63
  const int z = blockIdx.z;      // 0 = mu, 1 = logstd
  const float* W = z ? Ws : Wu;
  const float* bvec = z ? bs : bu;
  const int n = lane & 15;
  const int koff = (lane >> 4) * 2;
  const int nc = ntile * 16 + n;

  const float bias = bvec[nc];
  v8f acc;
#pragma unroll
  for (int v = 0; v < 8; ++v) acc[v] = bias;

  const float* hrow = h1 + (size_t)(mtile * 16 + n) * HH + koff;
#pragma unroll 8
  for (int kb = 0; kb < HH / 4; ++kb) {
    v2f a = *(const v2f*)(hrow + kb * 4);
    v2f b;
    b.x = W[(size_t)(kb * 4 + koff + 0) * DD + nc];
    b.y = W[(size_t)(kb * 4 + koff + 1) * DD + nc];
    acc = WMMA_F32(a, b, acc);
  }

  float* o = out + (size_t)z * BB * DD;
#pragma unroll
  for (int v = 0; v < 8; ++v) {
    const int m = v + (lane >> 4) * 8;
    o[(size_t)(mtile * 16 + m) * DD + nc] = acc[v];
  }
}

// ---------------------------------------------------------------------------
extern "C" void kernel_launch(void* const* d_in, const int* in_sizes, int n_in,
                              void* d_out, int out_size, void* d_ws,
                              size_t ws_size, hipStream_t stream) {
  const float* codes = (const float*)d_in[0];
  const float* table = (const float*)d_in[1];
  const float* W1 = (const float*)d_in[2];
  const float* b1 = (const float*)d_in[3];
  const float* Wu = (const float*)d_in[4];
  const float* bu = (const float*)d_in[5];
  const float* Ws = (const float*)d_in[6];
  const float* bs = (const float*)d_in[7];
  float* out = (float*)d_out;

  // Workspace layout (bytes)
  char* ws = (char*)d_ws;
  float* tnorm = (float*)(ws);                     // N*4        = 400000
  float* pval  = (float*)(ws + 400000);            // 64*64*16*4 = 262144
  int*   pidx  = (int*)  (ws + 400000 + 262144);   //            = 262144
  int*   nnidx = (int*)  (ws + 400000 + 524288);   // 1024*4     = 4096
  float* h1    = (float*)(ws + 400000 + 524288 + 4096);  // 1024*512*4 = 2 MB

  tnorm_kernel<<<(NN + 255) / 256, 256, 0, stream>>>(table, tnorm);
  nn_kernel<<<dim3(BB / 16 / RTILES, NSPLIT), 32, 0, stream>>>(
      codes, table, tnorm, pval, pidx);
  reduce_kernel<<<(BB + 255) / 256, 256, 0, stream>>>(pval, pidx, nnidx);
  h1_kernel<<<dim3(HH / 16, BB / 16), 32, 0, stream>>>(table, nnidx, W1, b1,
                                                       h1);
  out_kernel<<<dim3(DD / 16, BB / 16, 2), 32, 0, stream>>>(h1, Wu, bu, Ws, bs,
                                                           out);
}